// LinearAttention_75153337745549
// MI455X (gfx1250) — compile-verified
//
#include <hip/hip_runtime.h>
#include <cstdint>

// ---------------------------------------------------------------------------
// Problem constants (B,N,C,H,D from the reference)
// ---------------------------------------------------------------------------
#define B_ 4
#define N_ 1024
#define C_ 768
#define H_ 12
#define D_ 64
#define SCALE_ 0.125f            // D^-0.5 = 64^-0.5

typedef __attribute__((ext_vector_type(2))) float v2f;
typedef __attribute__((ext_vector_type(8))) float v8f;

// Per-matrix element counts in the workspace
#define S1_ ((size_t)B_ * H_ * N_ * D_)   // 3,145,728 floats = 12.6 MB

// fp32 WMMA: D(16x16,f32) = A(16x4,f32) * B(4x16,f32) + C
__device__ __forceinline__ v8f wmma4(v2f a, v2f b, v8f c) {
  return __builtin_amdgcn_wmma_f32_16x16x4_f32(
      /*neg_a=*/false, a, /*neg_b=*/false, b,
      /*c_mod=*/(short)0, c, /*reuse_a=*/false, /*reuse_b=*/false);
}

#define ZERO8 {0.f,0.f,0.f,0.f,0.f,0.f,0.f,0.f}

// CDNA5 async memory->LDS copy (no VGPR round trip, tracked by ASYNCcnt).
// The low 32 bits of a generic pointer to LDS are the LDS byte offset
// (ISA 10.2, LDS aperture).
__device__ __forceinline__ void async_copy_b128(float* lds_dst, const float* gsrc) {
  const uint32_t l = (uint32_t)(uintptr_t)lds_dst;
  asm volatile("global_load_async_to_lds_b128 %0, %1, off"
               :: "v"(l), "v"(gsrc)
               : "memory");
}
__device__ __forceinline__ void wait_async0() {
  asm volatile("s_wait_asynccnt 0" ::: "memory");
}

// ---------------------------------------------------------------------------
// Kernel 1: qkv = [X; Xj] @ W_qkv^T, scattered into q,k,v / qj,kj,vj
// One wave per 32x16 output tile for BOTH X and Xj (4 accumulators) so each
// W B-fragment load feeds 4 WMMAs (quarters the L2 weight traffic).
// grid = (BN/32 = 128, 3C/16 = 144), block = 32 (one wave)
// ---------------------------------------------------------------------------
__global__ __launch_bounds__(32)
void qkv_gemm_k(const float* __restrict__ X, const float* __restrict__ Xj,
                const float* __restrict__ W, float* __restrict__ ws)
{
  const int lane = threadIdx.x;
  const int hi   = lane >> 4;        // K-half select (ISA fp32 A/B layout)
  const int lr   = lane & 15;
  const int tm   = blockIdx.x;       // 32-row tile over B*N
  const int n0   = blockIdx.y * 16;  // output column base in [0, 2304)

  const float* xrow0  = X  + (size_t)(tm * 32 + lr) * C_;
  const float* xrow1  = X  + (size_t)(tm * 32 + 16 + lr) * C_;
  const float* xjrow0 = Xj + (size_t)(tm * 32 + lr) * C_;
  const float* xjrow1 = Xj + (size_t)(tm * 32 + 16 + lr) * C_;
  const float* wrow   = W  + (size_t)(n0 + lr) * C_;

  v8f accA0 = ZERO8, accA1 = ZERO8;  // primal tiles
  v8f accJ0 = ZERO8, accJ1 = ZERO8;  // tangent tiles
#pragma unroll 8
  for (int kk = 0; kk < C_ / 4; ++kk) {
    const int k0 = 4 * kk + 2 * hi;
    v2f b   = *(const v2f*)(wrow   + k0);
    v2f a0  = *(const v2f*)(xrow0  + k0);
    v2f a1  = *(const v2f*)(xrow1  + k0);
    v2f aj0 = *(const v2f*)(xjrow0 + k0);
    v2f aj1 = *(const v2f*)(xjrow1 + k0);
    accA0 = wmma4(a0,  b, accA0);
    accA1 = wmma4(a1,  b, accA1);
    accJ0 = wmma4(aj0, b, accJ0);
    accJ1 = wmma4(aj1, b, accJ1);
  }

  // Decode output column -> (qkv-select s, head h, dim d); uniform per block
  const int s  = n0 / C_;            // 0=q, 1=k, 2=v
  const int rm = n0 % C_;
  const int h  = rm / D_;
  const int d  = (rm % D_) + lr;
  const float scale = (s == 0) ? SCALE_ : 1.0f;   // q and q_jvp pre-scaled

  float* dstA = ws + (size_t)s       * S1_;       // q/k/v
  float* dstJ = ws + (size_t)(3 + s) * S1_;       // qj/kj/vj
#pragma unroll
  for (int i = 0; i < 8; ++i) {
    const int m0 = tm * 32 + i + 8 * hi;          // rows of first tile
    const int m1 = m0 + 16;                       // rows of second tile
    const int b0 = m0 >> 10, nn0 = m0 & (N_ - 1);
    const int b1 = m1 >> 10, nn1 = m1 & (N_ - 1);
    const size_t o0 = ((((size_t)b0 * H_ + h) << 10) + nn0) * D_ + d;
    const size_t o1 = ((((size_t)b1 * H_ + h) << 10) + nn1) * D_ + d;
    dstA[o0] = accA0[i] * scale;
    dstA[o1] = accA1[i] * scale;
    dstJ[o0] = accJ0[i] * scale;
    dstJ[o1] = accJ1[i] * scale;
  }
}

// ---------------------------------------------------------------------------
// Kernel 2: fused attention forward + JVP for one (b,h) and one 16-row q tile
// 8 waves / 256 threads per workgroup, 192 KB dynamic LDS (CDNA5 320KB WGP).
//   LDS: sS[16][1024] | sDm[16][1024] | sA[128][64] | sB[128][64]
// sA/sB stage k/kj in phase 1 (async copies), reductions in phase 2,
// and v/vj in phase 3 (async copies).
// grid = (N/16 = 64, B*H = 48), block = 256
// ---------------------------------------------------------------------------
__global__ __launch_bounds__(256)
void attn_jvp_k(float* __restrict__ ws)
{
  extern __shared__ float smem[];
  float* sS  = smem;                        // 16384 floats
  float* sDm = smem + 16 * 1024;            // 16384 floats
  float* sA  = smem + 32 * 1024;            // 8192 floats (k, red, v)
  float* sB  = smem + 32 * 1024 + 128 * D_; // 8192 floats (kj, vj)

  const int tid  = threadIdx.x;
  const int lane = tid & 31;
  const int w    = tid >> 5;               // wave 0..7
  const int hi   = lane >> 4;
  const int lr   = lane & 15;

  const int bh    = blockIdx.y;            // 0..47
  const int rbase = blockIdx.x * 16;       // q-row tile base

  const float* qg  = ws + 0 * S1_ + (size_t)bh * N_ * D_;
  const float* kg  = ws + 1 * S1_ + (size_t)bh * N_ * D_;
  const float* vg  = ws + 2 * S1_ + (size_t)bh * N_ * D_;
  const float* qjg = ws + 3 * S1_ + (size_t)bh * N_ * D_;
  const float* kjg = ws + 4 * S1_ + (size_t)bh * N_ * D_;
  const float* vjg = ws + 5 * S1_ + (size_t)bh * N_ * D_;
  float* xb  = ws + 6 * S1_;               // [B][N][C] for proj
  float* xjb = ws + 7 * S1_;

  // --- Preload q / q_jvp A-fragments for the 16-row tile (K = 0..63) -------
  v2f qf[16], qjf[16];
  {
    const float* qrow  = qg  + (size_t)(rbase + lr) * D_;
    const float* qjrow = qjg + (size_t)(rbase + lr) * D_;
#pragma unroll
    for (int kk = 0; kk < 16; ++kk) {
      qf[kk]  = *(const v2f*)(qrow  + 4 * kk + 2 * hi);
      qjf[kk] = *(const v2f*)(qjrow + 4 * kk + 2 * hi);
    }
  }

  // --- Phase 1: S = q k^T, dm = qj k^T + q kj^T, into LDS row buffers ------
  for (int t = 0; t < 8; ++t) {            // 128-column super tiles
    {
      const float* ksrc  = kg  + (size_t)(t * 128) * D_;
      const float* kjsrc = kjg + (size_t)(t * 128) * D_;
#pragma unroll
      for (int i = 0; i < 8; ++i) {        // 8192 floats per array, 256 thr
        const int idx = (i * 256 + tid) * 4;
        async_copy_b128(sA + idx, ksrc  + idx);   // global -> LDS, no VGPRs
        async_copy_b128(sB + idx, kjsrc + idx);
      }
      wait_async0();
    }
    __syncthreads();

    const int ncol = w * 16;               // this wave's local 16 columns
    const float* krow  = sA + (size_t)(ncol + lr) * D_;
    const float* kjrow = sB + (size_t)(ncol + lr) * D_;
    v8f sacc = ZERO8;
    v8f dacc = ZERO8;
#pragma unroll
    for (int kk = 0; kk < 16; ++kk) {
      v2f bk  = *(const v2f*)(krow  + 4 * kk + 2 * hi);
      v2f bkj = *(const v2f*)(kjrow + 4 * kk + 2 * hi);
      sacc = wmma4(qf[kk],  bk,  sacc);
      dacc = wmma4(qjf[kk], bk,  dacc);    // A_jvp
      dacc = wmma4(qf[kk],  bkj, dacc);    // + B_jvp
    }
    const int col = t * 128 + ncol + lr;
#pragma unroll
    for (int i = 0; i < 8; ++i) {
      sS [(i + 8 * hi) * 1024 + col] = sacc[i];
      sDm[(i + 8 * hi) * 1024 + col] = dacc[i];
    }
    __syncthreads();
  }

  // --- Phase 2: softmax + JVP filter (row ops, 16 threads per row) ---------
  {
    float* red = sA;                       // staging space is free now
    const int r = tid >> 4, c = tid & 15;
    float* srow = sS  + r * 1024;
    float* drow = sDm + r * 1024;

    float mx = -3.402823466e38f;
#pragma unroll 8
    for (int j = 0; j < 64; ++j) mx = fmaxf(mx, srow[c + 16 * j]);
    red[tid] = mx;
    __syncthreads();
    if (c == 0) {
      float m2 = red[r * 16];
      for (int j = 1; j < 16; ++j) m2 = fmaxf(m2, red[r * 16 + j]);
      red[512 + r] = m2;
    }
    __syncthreads();

    const float m2 = red[512 + r];
    float se = 0.f, sd = 0.f;
#pragma unroll 8
    for (int j = 0; j < 64; ++j) {
      const int idx = c + 16 * j;
      const float e = __expf(srow[idx] - m2);
      srow[idx] = e;
      se += e;
      sd += drow[idx] * e;                 // rowsum(dm * exp)
    }
    red[tid] = se;
    red[256 + tid] = sd;
    __syncthreads();
    if (c == 0) {
      float sum = 0.f, dot = 0.f;
      for (int j = 0; j < 16; ++j) { sum += red[r * 16 + j]; dot += red[256 + r * 16 + j]; }
      const float inv = 1.0f / sum;
      red[544 + r] = inv;
      red[576 + r] = dot * inv;            // rowdot = sum(dm * attn)
    }
    __syncthreads();

    const float inv = red[544 + r];
    const float rd  = red[576 + r];
#pragma unroll 8
    for (int j = 0; j < 64; ++j) {
      const int idx = c + 16 * j;
      const float a  = srow[idx] * inv;    // attn
      const float dm = drow[idx];
      srow[idx] = a;                       // sS  <- attn
      drow[idx] = a * (dm - rd);           // sDm <- attn_filter
    }
  }
  __syncthreads();

  // --- Phase 3: x = attn@v ; x_jvp = attn@vj + filter@v (K = 1024) ---------
  {
    const int which = w >> 2;              // 0 -> x, 1 -> x_jvp
    const int d0    = (w & 3) * 16;        // 16-wide slice of D
    float* sV  = sA;                       // reuse staging space for v
    float* sVj = sB;                       // and v_jvp
    v8f acc = ZERO8;

    for (int t = 0; t < 8; ++t) {          // 128-key super tiles
      {
        const float* vsrc  = vg  + (size_t)(t * 128) * D_;
        const float* vjsrc = vjg + (size_t)(t * 128) * D_;
#pragma unroll
        for (int i = 0; i < 8; ++i) {
          const int idx = (i * 256 + tid) * 4;
          async_copy_b128(sV  + idx, vsrc  + idx);
          async_copy_b128(sVj + idx, vjsrc + idx);
        }
        wait_async0();
      }
      __syncthreads();

      const float* arow = sS  + lr * 1024 + t * 128;  // attn rows
      const float* frow = sDm + lr * 1024 + t * 128;  // filter rows
#pragma unroll 4
      for (int kk = 0; kk < 32; ++kk) {    // 128 K values per super tile
        const int kl = 4 * kk + 2 * hi;
        v2f aa = *(const v2f*)(arow + kl);
        v2f bv = { sV[kl * D_ + d0 + lr],
                   sV[(kl + 1) * D_ + d0 + lr] };
        if (which == 0) {
          acc = wmma4(aa, bv, acc);
        } else {
          v2f ff  = *(const v2f*)(frow + kl);
          v2f bvj = { sVj[kl * D_ + d0 + lr],
                      sVj[(kl + 1) * D_ + d0 + lr] };
          acc = wmma4(aa, bvj, acc);       // attn @ v_jvp
          acc = wmma4(ff, bv,  acc);       // + filter @ v
        }
      }
      __syncthreads();
    }

    const int b = bh / H_, h = bh % H_;
    float* dst = (which == 0) ? xb : xjb;
#pragma unroll
    for (int i = 0; i < 8; ++i) {
      const int n = rbase + i + 8 * hi;
      dst[((size_t)b * N_ + n) * C_ + h * D_ + d0 + lr] = acc[i];
    }
  }
}

// ---------------------------------------------------------------------------
// Kernel 3: out = x @ W_proj^T + b ; out_jvp = x_jvp @ W_proj^T
// Same 32-row M-blocking as kernel 1 for W reuse.
// grid = (BN/32 = 128, C/16 = 48), block = 32 (one wave)
// ---------------------------------------------------------------------------
__global__ __launch_bounds__(32)
void proj_gemm_k(const float* __restrict__ ws, const float* __restrict__ W,
                 const float* __restrict__ bias, float* __restrict__ out)
{
  const int lane = threadIdx.x;
  const int hi   = lane >> 4;
  const int lr   = lane & 15;
  const int tm   = blockIdx.x;
  const int n0   = blockIdx.y * 16;

  const float* xrow0  = ws + 6 * S1_ + (size_t)(tm * 32 + lr) * C_;
  const float* xrow1  = ws + 6 * S1_ + (size_t)(tm * 32 + 16 + lr) * C_;
  const float* xjrow0 = ws + 7 * S1_ + (size_t)(tm * 32 + lr) * C_;
  const float* xjrow1 = ws + 7 * S1_ + (size_t)(tm * 32 + 16 + lr) * C_;
  const float* wrow   = W + (size_t)(n0 + lr) * C_;

  v8f accA0 = ZERO8, accA1 = ZERO8;
  v8f accJ0 = ZERO8, accJ1 = ZERO8;
#pragma unroll 8
  for (int kk = 0; kk < C_ / 4; ++kk) {
    const int k0 = 4 * kk + 2 * hi;
    v2f b   = *(const v2f*)(wrow   + k0);
    v2f a0  = *(const v2f*)(xrow0  + k0);
    v2f a1  = *(const v2f*)(xrow1  + k0);
    v2f aj0 = *(const v2f*)(xjrow0 + k0);
    v2f aj1 = *(const v2f*)(xjrow1 + k0);
    accA0 = wmma4(a0,  b, accA0);
    accA1 = wmma4(a1,  b, accA1);
    accJ0 = wmma4(aj0, b, accJ0);
    accJ1 = wmma4(aj1, b, accJ1);
  }

  const float bval = bias[n0 + lr];
  const size_t half = (size_t)B_ * N_ * C_;
#pragma unroll
  for (int i = 0; i < 8; ++i) {
    const int m0 = tm * 32 + i + 8 * hi;        // row in B*N
    const int m1 = m0 + 16;
    out[(size_t)m0 * C_ + n0 + lr]        = accA0[i] + bval;
    out[(size_t)m1 * C_ + n0 + lr]        = accA1[i] + bval;
    out[half + (size_t)m0 * C_ + n0 + lr] = accJ0[i];
    out[half + (size_t)m1 * C_ + n0 + lr] = accJ1[i];
  }
}

// ---------------------------------------------------------------------------
// Host launcher
// ---------------------------------------------------------------------------
extern "C" void kernel_launch(void* const* d_in, const int* in_sizes, int n_in,
                              void* d_out, int out_size, void* d_ws, size_t ws_size,
                              hipStream_t stream) {
  const float* X  = (const float*)d_in[0];   // [B,N,C]
  const float* Xj = (const float*)d_in[1];   // [B,N,C]
  const float* Wq = (const float*)d_in[2];   // [3C,C]
  const float* Wp = (const float*)d_in[3];   // [C,C]
  const float* bp = (const float*)d_in[4];   // [C]
  float* out = (float*)d_out;
  float* ws  = (float*)d_ws;                 // needs 8 * S1_ floats (~101 MB)

  dim3 g1((B_ * N_) / 32, (3 * C_) / 16);
  qkv_gemm_k<<<g1, 32, 0, stream>>>(X, Xj, Wq, ws);

  dim3 g2(N_ / 16, B_ * H_);
  const size_t lds = (size_t)(2 * 16 * 1024 + 2 * 128 * D_) * sizeof(float); // 192 KB
  attn_jvp_k<<<g2, 256, lds, stream>>>(ws);

  dim3 g3((B_ * N_) / 32, C_ / 16);
  proj_gemm_k<<<g3, 32, 0, stream>>>(ws, Wp, bp, out);
}